// PointFP_21586505630014
// MI455X (gfx1250) — compile-verified
//
#include <hip/hip_runtime.h>

typedef __attribute__((ext_vector_type(2))) float v2f;
typedef __attribute__((ext_vector_type(8))) float v8f;

constexpr int B_   = 2;
constexpr int G_   = 16;
constexpr int N1_  = 4096;
constexpr int S1_  = 1024;
constexpr int D1_  = 128;
constexpr int D2_  = 256;
constexpr int INCH = D1_ + D2_;   // 384
constexpr int C1   = 256;
constexpr int C2   = 256;
constexpr int NPTS = G_ * N1_;    // 65536
constexpr int SPTS = G_ * S1_;    // 16384
constexpr float EPS_DIST = 1e-8f;
constexpr float EPS_BN   = 1e-5f;

// ---------------------------------------------------------------------------
// 1) 3-NN + inverse-distance weights. One block = 256 queries of one segment;
//    the 1024 source points live in LDS (12 KB).
// ---------------------------------------------------------------------------
__global__ void __launch_bounds__(256)
knn_kernel(const float* __restrict__ xyz1, const float* __restrict__ xyz2,
           int* __restrict__ knn_idx, float* __restrict__ knn_w) {
  __shared__ float sx[S1_], sy[S1_], sz[S1_];
  const int blk = blockIdx.x;       // B*G*16 blocks
  const int qc  = blk & 15;
  const int bg  = blk >> 4;
  const int b = bg / G_, g = bg % G_;

  const float* x2 = xyz2 + (size_t)b * 3 * SPTS + g * S1_;
  for (int j = threadIdx.x; j < S1_; j += 256) {
    sx[j] = x2[j];
    sy[j] = x2[SPTS + j];
    sz[j] = x2[2 * SPTS + j];
  }
  __syncthreads();

  const int q = qc * 256 + threadIdx.x;
  const float* x1 = xyz1 + (size_t)b * 3 * NPTS + g * N1_ + q;
  const float px = x1[0], py = x1[NPTS], pz = x1[2 * NPTS];

  float d0 = 3.4e38f, d1 = 3.4e38f, d2v = 3.4e38f;
  int   i0 = 0, i1 = 0, i2 = 0;
  for (int j = 0; j < S1_; ++j) {
    const float dx = px - sx[j], dy = py - sy[j], dz = pz - sz[j];
    const float d = fmaf(dx, dx, fmaf(dy, dy, dz * dz));
    if (d < d2v) {
      if (d < d1) {
        d2v = d1; i2 = i1;
        if (d < d0) { d1 = d0; i1 = i0; d0 = d; i0 = j; }
        else        { d1 = d;  i1 = j; }
      } else { d2v = d; i2 = j; }
    }
  }
  const float w0 = 1.f / (d0 + EPS_DIST);
  const float w1 = 1.f / (d1 + EPS_DIST);
  const float w2 = 1.f / (d2v + EPS_DIST);
  const float inv = 1.f / (w0 + w1 + w2);
  const size_t base = ((size_t)bg * N1_ + q) * 3;
  knn_idx[base + 0] = i0; knn_idx[base + 1] = i1; knn_idx[base + 2] = i2;
  knn_w[base + 0] = w0 * inv; knn_w[base + 1] = w1 * inv; knn_w[base + 2] = w2 * inv;
}

// ---------------------------------------------------------------------------
// 2) Transpose weights to k-major so GEMM LDS staging is coalesced.
// ---------------------------------------------------------------------------
__global__ void __launch_bounds__(256)
transpose_kernel(const float* __restrict__ W, float* __restrict__ Wt, int M, int K) {
  const int i = blockIdx.x * 256 + threadIdx.x;
  if (i >= M * K) return;
  const int o = i / K, k = i % K;
  Wt[k * M + o] = W[i];
}

// ---------------------------------------------------------------------------
// 3) WMMA GEMM: Out[b][0:256][seg cols] = Wt^T @ Htile.
//    MODE 0: H = concat(points1, 3-NN-interp(points2))  (interp fused in staging)
//    MODE 1: H = relu(bn1(O1))                          (BN fused in staging)
//    f32 16x16x4 WMMA; block = 256 rows x 64 cols, 8 waves x (2M x 4N) tiles.
//    LDS tiles are stored k-PAIR-major (v2f per element) so every A/B fragment
//    is one aligned ds_load_b64 straight into the WMMA operand pair.
// ---------------------------------------------------------------------------
constexpr int NT    = 64;        // columns per block
constexpr int KC    = 32;        // K chunk staged in LDS (16 k-pairs)
constexpr int KP    = KC / 2;    // k-pairs per chunk
constexpr int HPAD2 = NT + 2;    // v2f row stride (pad: de-alias half-wave rows)
constexpr int WPAD2 = 256 + 2;   // v2f row stride

template <int KTOT, int MODE>
__global__ void __launch_bounds__(256)
gemm_kernel(const float* __restrict__ Wt,        // [KTOT][256] k-major
            const float* __restrict__ p1,        // MODE0: points1
            const float* __restrict__ p2,        // MODE0: points2
            const int*   __restrict__ knn_idx,   // MODE0
            const float* __restrict__ knn_w,     // MODE0
            const float* __restrict__ X,         // MODE1: raw O1
            const float* __restrict__ scale,     // MODE1: [G][256]
            const float* __restrict__ shift,     // MODE1: [G][256]
            float* __restrict__ Out) {           // raw [B][256][NPTS]
  __shared__ v2f Wl[KP * WPAD2];   // {k even, k odd} per element
  __shared__ v2f Hl[KP * HPAD2];
  __shared__ int   sIdx[NT * 3];
  __shared__ float sW[NT * 3];

  const int blk   = blockIdx.x;
  const int ntile = blk % (N1_ / NT);
  const int bg    = blk / (N1_ / NT);
  const int b = bg / G_, g = bg % G_;
  const int n0  = ntile * NT;
  const int tid = threadIdx.x;
  const int lane = tid & 31, wv = tid >> 5;
  const int mbase = wv * 32;

  if (MODE == 0) {
    for (int e = tid; e < NT * 3; e += 256) {
      const size_t base = ((size_t)bg * N1_ + n0) * 3 + e;
      sIdx[e] = knn_idx[base];
      sW[e]   = knn_w[base];
    }
  }

  const v8f vzero = {0.f, 0.f, 0.f, 0.f, 0.f, 0.f, 0.f, 0.f};
  v8f acc[2][4];
#pragma unroll
  for (int mt = 0; mt < 2; ++mt)
#pragma unroll
    for (int nt = 0; nt < 4; ++nt) acc[mt][nt] = vzero;

  // Compute one activation element for channel ch, column n (within the block's 64-column tile).
  auto hval = [&](int ch, int n) -> float {
    if (MODE == 0) {
      if (ch < D1_) {
        return p1[(size_t)b * D1_ * NPTS + (size_t)ch * NPTS + g * N1_ + n0 + n];
      }
      const int c2 = ch - D1_;
      const float* row = p2 + (size_t)b * D2_ * SPTS + (size_t)c2 * SPTS + g * S1_;
      float v = sW[n * 3 + 0] * row[sIdx[n * 3 + 0]];
      v = fmaf(sW[n * 3 + 1], row[sIdx[n * 3 + 1]], v);
      v = fmaf(sW[n * 3 + 2], row[sIdx[n * 3 + 2]], v);
      return v;
    } else {
      const float x = X[(size_t)b * C1 * NPTS + (size_t)ch * NPTS + g * N1_ + n0 + n];
      return fmaxf(fmaf(x, scale[g * C1 + ch], shift[g * C1 + ch]), 0.f);
    }
  };

  for (int kb = 0; kb < KTOT; kb += KC) {
    __syncthreads();
    // Stage weight chunk as k-pairs (coalesced: k-major source).
    for (int e = tid; e < KP * 256; e += 256) {
      const int kp = e >> 8, m = e & 255;
      v2f w;
      w.x = Wt[(size_t)(kb + 2 * kp + 0) * 256 + m];
      w.y = Wt[(size_t)(kb + 2 * kp + 1) * 256 + m];
      Wl[kp * WPAD2 + m] = w;
    }
    if (kb + KC < KTOT)
      __builtin_prefetch(&Wt[(size_t)(kb + KC) * 256 + tid], 0, 0);
    // Stage activation chunk as k-pairs, with interp / BN fused.
    for (int e = tid; e < KP * NT; e += 256) {
      const int kp = e / NT, n = e % NT;
      v2f h;
      h.x = hval(kb + 2 * kp + 0, n);
      h.y = hval(kb + 2 * kp + 1, n);
      Hl[kp * HPAD2 + n] = h;
    }
    __syncthreads();

    // f32 A/B frag: lanes 0-15 hold K={0,1}, lanes 16-31 hold K={2,3}
    // -> pair row = ks + (lane>=16).  Single ds_load_b64 per fragment.
    const int khi = lane >> 4;
    const int ml  = lane & 15;
#pragma unroll
    for (int ks = 0; ks < KP; ks += 2) {     // ks counts k-pairs (K step 4)
      const int prow = ks + khi;
      v2f bf[4];
#pragma unroll
      for (int nt = 0; nt < 4; ++nt)
        bf[nt] = Hl[prow * HPAD2 + nt * 16 + ml];
#pragma unroll
      for (int mt = 0; mt < 2; ++mt) {
        const v2f af = Wl[prow * WPAD2 + mbase + mt * 16 + ml];
#pragma unroll
        for (int nt = 0; nt < 4; ++nt) {
          acc[mt][nt] = __builtin_amdgcn_wmma_f32_16x16x4_f32(
              false, af, false, bf[nt], (short)0, acc[mt][nt], false, false);
        }
      }
    }
  }

  // Write raw (pre-BN) result. C/D layout: VGPR v -> rows v (lanes 0-15) / v+8.
  const int ml  = lane & 15;
  const int mhi = (lane >> 4) * 8;
#pragma unroll
  for (int mt = 0; mt < 2; ++mt)
#pragma unroll
    for (int nt = 0; nt < 4; ++nt)
#pragma unroll
      for (int v = 0; v < 8; ++v) {
        const int m = mbase + mt * 16 + v + mhi;
        const int n = n0 + nt * 16 + ml;
        Out[(size_t)b * C1 * NPTS + (size_t)m * NPTS + g * N1_ + n] = acc[mt][nt][v];
      }
}

// ---------------------------------------------------------------------------
// 4) Per-(group, channel) BN stats over (B, n); prefold gamma/beta:
//    scale = gamma*rstd, shift = beta - mu*scale.  (Linear bias cancels in BN.)
// ---------------------------------------------------------------------------
__global__ void __launch_bounds__(256)
stats_kernel(const float* __restrict__ X, const float* __restrict__ gamma,
             const float* __restrict__ beta, float* __restrict__ scale,
             float* __restrict__ shift) {
  const int g = blockIdx.x / C1;
  const int o = blockIdx.x % C1;
  __shared__ float ssum[256], ssq[256];
  float s = 0.f, s2 = 0.f;
  for (int b = 0; b < B_; ++b) {
    const float* p = X + (size_t)b * C1 * NPTS + (size_t)o * NPTS + g * N1_;
    for (int n = threadIdx.x; n < N1_; n += 256) {
      const float x = p[n];
      s += x;
      s2 = fmaf(x, x, s2);
    }
  }
  ssum[threadIdx.x] = s;
  ssq[threadIdx.x]  = s2;
  __syncthreads();
  for (int st = 128; st > 0; st >>= 1) {
    if (threadIdx.x < st) {
      ssum[threadIdx.x] += ssum[threadIdx.x + st];
      ssq[threadIdx.x]  += ssq[threadIdx.x + st];
    }
    __syncthreads();
  }
  if (threadIdx.x == 0) {
    const float inv = 1.f / (float)(B_ * N1_);
    const float mu  = ssum[0] * inv;
    const float var = ssq[0] * inv - mu * mu;
    const float rs  = rsqrtf(var + EPS_BN);
    const float sc  = gamma[o] * rs;
    scale[blockIdx.x] = sc;                 // index == g*C1 + o
    shift[blockIdx.x] = beta[o] - mu * sc;
  }
}

// ---------------------------------------------------------------------------
// 5) Final BN2 + ReLU in-place on d_out.
// ---------------------------------------------------------------------------
__global__ void __launch_bounds__(256)
bn_apply_kernel(float* __restrict__ Y, const float* __restrict__ scale,
                const float* __restrict__ shift) {
  const size_t i = (size_t)blockIdx.x * 256 + threadIdx.x;  // B*C2*NPTS total
  const int o = (int)((i / NPTS) % C2);
  const int g = (int)((i % NPTS) / N1_);
  const float x = Y[i];
  Y[i] = fmaxf(fmaf(x, scale[g * C2 + o], shift[g * C2 + o]), 0.f);
}

// ---------------------------------------------------------------------------
extern "C" void kernel_launch(void* const* d_in, const int* in_sizes, int n_in,
                              void* d_out, int out_size, void* d_ws, size_t ws_size,
                              hipStream_t stream) {
  const float* xyz1    = (const float*)d_in[0];
  const float* points1 = (const float*)d_in[1];
  const float* xyz2    = (const float*)d_in[3];
  const float* points2 = (const float*)d_in[4];
  const float* W1      = (const float*)d_in[6];
  const float* gamma1  = (const float*)d_in[8];
  const float* beta1   = (const float*)d_in[9];
  const float* W2      = (const float*)d_in[10];
  const float* gamma2  = (const float*)d_in[12];
  const float* beta2   = (const float*)d_in[13];
  float* out = (float*)d_out;

  char* ws = (char*)d_ws;
  size_t off = 0;
  auto walloc = [&](size_t bytes) -> void* {
    void* p = ws + off;
    off = (off + bytes + 255) & ~(size_t)255;
    return p;
  };
  int*   knn_idx = (int*)  walloc(sizeof(int)   * (size_t)B_ * G_ * N1_ * 3);
  float* knn_w   = (float*)walloc(sizeof(float) * (size_t)B_ * G_ * N1_ * 3);
  float* Wt1     = (float*)walloc(sizeof(float) * (size_t)INCH * C1);
  float* Wt2     = (float*)walloc(sizeof(float) * (size_t)C1 * C2);
  float* scale1  = (float*)walloc(sizeof(float) * (size_t)G_ * C1);
  float* shift1  = (float*)walloc(sizeof(float) * (size_t)G_ * C1);
  float* scale2  = (float*)walloc(sizeof(float) * (size_t)G_ * C2);
  float* shift2  = (float*)walloc(sizeof(float) * (size_t)G_ * C2);
  float* O1      = (float*)walloc(sizeof(float) * (size_t)B_ * C1 * NPTS);

  knn_kernel<<<B_ * G_ * 16, 256, 0, stream>>>(xyz1, xyz2, knn_idx, knn_w);
  transpose_kernel<<<(C1 * INCH + 255) / 256, 256, 0, stream>>>(W1, Wt1, C1, INCH);
  transpose_kernel<<<(C2 * C1 + 255) / 256, 256, 0, stream>>>(W2, Wt2, C2, C1);

  gemm_kernel<INCH, 0><<<B_ * G_ * (N1_ / NT), 256, 0, stream>>>(
      Wt1, points1, points2, knn_idx, knn_w, nullptr, nullptr, nullptr, O1);

  stats_kernel<<<G_ * C1, 256, 0, stream>>>(O1, gamma1, beta1, scale1, shift1);

  gemm_kernel<C1, 1><<<B_ * G_ * (N1_ / NT), 256, 0, stream>>>(
      Wt2, nullptr, nullptr, nullptr, nullptr, O1, scale1, shift1, out);

  stats_kernel<<<G_ * C2, 256, 0, stream>>>(out, gamma2, beta2, scale2, shift2);
  bn_apply_kernel<<<((size_t)B_ * C2 * NPTS) / 256, 256, 0, stream>>>(out, scale2, shift2);
}